// Ext_relLayer_4561255268669
// MI455X (gfx1250) — compile-verified
//
#include <hip/hip_runtime.h>

typedef float v2f __attribute__((ext_vector_type(2)));
typedef float v8f __attribute__((ext_vector_type(8)));

#define N_NODES 50000
#define N_EDGES 640000
#define N_REL   500
#define DIM     128
#define K2      256   // 2*DIM

// d_ws float layout
#define WS_WT_I 0
#define WS_WT_O 32768              // 256*128
#define WS_WT_S 65536              // +256*128
#define WS_WT_M 81920              // +128*128
#define WS_AGG  98304              // +128*128
#define WS_CNT  6498304            // +50000*128
#define WS_ZERO_N 6450000          // agg + cnt floats to zero

#define A_STRIDE_E 260             // padded 256: %4==0 (b128 stores), %64==4 (no bank conflicts)
#define A_STRIDE_N 132             // padded 128: %4==0, %64==4

#define EDGE_MT    4               // M-tiles (of 16 edges) per wave
#define EDGE_WAVES 4               // waves per block (128 threads)
#define EDGES_PER_WAVE  (EDGE_MT * 16)            // 64
#define EDGES_PER_BLOCK (EDGE_WAVES * EDGES_PER_WAVE)  // 256

// ---------------------------------------------------------------- utilities
__global__ __launch_bounds__(256) void k_zero(float* __restrict__ p, int n) {
    int i = blockIdx.x * 256 + threadIdx.x;
    if (i < n) p[i] = 0.0f;
}

// src is [R][C] row-major, dst becomes [C][R] row-major (dst[c*R+r]=src[r*C+c])
__global__ __launch_bounds__(256) void k_transpose(const float* __restrict__ src,
                                                   float* __restrict__ dst,
                                                   int R, int C) {
    int i = blockIdx.x * 256 + threadIdx.x;
    if (i < R * C) {
        int r = i / C, c = i - r * C;
        dst[c * R + r] = src[i];
    }
}

// ---------------------------------------------------------------- edge GEMM
// per wave: 64 edges (4 WMMA M-tiles, register-blocked); per block: 256 edges.
// B fragments (WtI/WtO) loaded once per k-step and reused by all 4 M-tiles.
__global__ __launch_bounds__(128) void k_edge(const float* __restrict__ rel_emb,
                                              const float* __restrict__ metarel,
                                              const int*   __restrict__ g_src,
                                              const int*   __restrict__ g_dst,
                                              const int*   __restrict__ g_rel,
                                              const int*   __restrict__ g_inv,
                                              const float* __restrict__ WtI,
                                              const float* __restrict__ bI,
                                              const float* __restrict__ WtO,
                                              const float* __restrict__ bO,
                                              float* __restrict__ agg,
                                              float* __restrict__ cnt) {
    extern __shared__ float lds[];
    const int lane = threadIdx.x & 31;
    const int wave = threadIdx.x >> 5;
    const int lr   = lane & 15;
    const int half = lane >> 4;

    float* aw = lds + wave * (EDGES_PER_WAVE * A_STRIDE_E);
    const int e0 = (blockIdx.x * EDGE_WAVES + wave) * EDGES_PER_WAVE;

    // ---- stage comp_h = [metarel[rel[e]] | rel_emb[src[e]]]  (64 x 256)
    const int col = lane * 8;  // 32 lanes * 8 = 256 columns
    for (int e = 0; e < EDGES_PER_WAVE; ++e) {
        const int ed = e0 + e;
        const int rr = g_rel[ed];
        const int ss = g_src[ed];
        const float* p = (col < DIM) ? (metarel + rr * DIM + col)
                                     : (rel_emb + ss * DIM + (col - DIM));
        float4 x0 = *(const float4*)p;
        float4 x1 = *(const float4*)(p + 4);
        float* q = aw + e * A_STRIDE_E + col;   // dword offset %4==0 -> 16B aligned
        *(float4*)(q)     = x0;
        *(float4*)(q + 4) = x1;
    }
    // in-degree counts: 64 edges per wave, 2 per lane
    atomicAdd(&cnt[g_dst[e0 + lane]],      1.0f);
    atomicAdd(&cnt[g_dst[e0 + 32 + lane]], 1.0f);

    // ---- 8 N-tiles of 16 outputs; K = 256 -> 64 wmma k-steps; 8 WMMAs/k-step
    for (int nt = 0; nt < 8; ++nt) {
        const int n0 = nt * 16;
        v8f accI[EDGE_MT];
        v8f accO[EDGE_MT];
#pragma unroll
        for (int t = 0; t < EDGE_MT; ++t) { accI[t] = {}; accO[t] = {}; }

#pragma unroll 2
        for (int ks = 0; ks < 64; ++ks) {
            // A frags: lanes 0-15 K=4ks,4ks+1 ; lanes 16-31 K=4ks+2,4ks+3
            v2f a[EDGE_MT];
#pragma unroll
            for (int t = 0; t < EDGE_MT; ++t) {
                const int ao = (t * 16 + lr) * A_STRIDE_E + ks * 4 + half * 2;
                a[t].x = aw[ao];
                a[t].y = aw[ao + 1];
            }
            // B frags from transposed weights Wt[k][n] (shared by all M-tiles)
            const int bo0 = (ks * 4 + half * 2) * DIM + n0 + lr;
            v2f bi, bo;
            bi.x = WtI[bo0];  bi.y = WtI[bo0 + DIM];
            bo.x = WtO[bo0];  bo.y = WtO[bo0 + DIM];
#pragma unroll
            for (int t = 0; t < EDGE_MT; ++t) {
                accI[t] = __builtin_amdgcn_wmma_f32_16x16x4_f32(false, a[t], false, bi,
                                                                (short)0, accI[t], false, false);
                accO[t] = __builtin_amdgcn_wmma_f32_16x16x4_f32(false, a[t], false, bo,
                                                                (short)0, accO[t], false, false);
            }
        }
        const int n = n0 + lr;
        const float biasI = bI[n];
        const float biasO = bO[n];
#pragma unroll
        for (int t = 0; t < EDGE_MT; ++t) {
#pragma unroll
            for (int r = 0; r < 8; ++r) {
                const int ed = e0 + t * 16 + r + half * 8;
                const int iv = g_inv[ed];
                const int dm = g_dst[ed];
                const float v = iv ? (accO[t][r] + biasO) : (accI[t][r] + biasI);
                atomicAdd(&agg[dm * DIM + n], v);
            }
        }
    }
}

// ---------------------------------------------------------------- node GEMM
// rel_new = rel_emb @ W_S^T + b_S + agg / max(cnt,1)
__global__ __launch_bounds__(256) void k_node(const float* __restrict__ rel_emb,
                                              const float* __restrict__ WtS,
                                              const float* __restrict__ bS,
                                              const float* __restrict__ agg,
                                              const float* __restrict__ cnt,
                                              float* __restrict__ out) {
    extern __shared__ float lds[];
    const int lane = threadIdx.x & 31;
    const int wave = threadIdx.x >> 5;
    const int lr   = lane & 15;
    const int half = lane >> 4;

    float* aw = lds + wave * (16 * A_STRIDE_N);
    const int m0 = (blockIdx.x * 8 + wave) * 16;
    if (m0 >= N_NODES) return;   // 50000 % 16 == 0: tiles are all-or-nothing

    const int col = lane * 4;    // 32 lanes * 4 = 128
    for (int r = 0; r < 16; ++r) {
        float4 x = *(const float4*)(rel_emb + (m0 + r) * DIM + col);
        *(float4*)(aw + r * A_STRIDE_N + col) = x;
    }

    for (int nt = 0; nt < 8; ++nt) {
        const int n0 = nt * 16;
        v8f acc = {};
#pragma unroll 4
        for (int ks = 0; ks < 32; ++ks) {
            const int ao = lr * A_STRIDE_N + ks * 4 + half * 2;
            v2f a;
            a.x = aw[ao];
            a.y = aw[ao + 1];
            const int bo0 = (ks * 4 + half * 2) * DIM + n0 + lr;
            v2f b;
            b.x = WtS[bo0];
            b.y = WtS[bo0 + DIM];
            acc = __builtin_amdgcn_wmma_f32_16x16x4_f32(false, a, false, b,
                                                        (short)0, acc, false, false);
        }
        const int n = n0 + lr;
        const float bias = bS[n];
#pragma unroll
        for (int r = 0; r < 8; ++r) {
            const int m = m0 + r + half * 8;
            const float c  = cnt[m];
            const float ha = agg[m * DIM + n] / fmaxf(c, 1.0f);
            out[m * DIM + n] = acc[r] + bias + ha;
        }
    }
}

// ---------------------------------------------------------------- meta GEMM
// meta_new = metarel @ W_M^T + b_M   (500 rows, zero-padded partial tile)
__global__ __launch_bounds__(256) void k_meta(const float* __restrict__ metarel,
                                              const float* __restrict__ WtM,
                                              const float* __restrict__ bM,
                                              float* __restrict__ out) {
    extern __shared__ float lds[];
    const int lane = threadIdx.x & 31;
    const int wave = threadIdx.x >> 5;
    const int lr   = lane & 15;
    const int half = lane >> 4;

    float* aw = lds + wave * (16 * A_STRIDE_N);
    const int m0 = (blockIdx.x * 8 + wave) * 16;
    if (m0 >= N_REL) return;

    const int col = lane * 4;
    for (int r = 0; r < 16; ++r) {
        const int m = m0 + r;
        float4 x = make_float4(0.f, 0.f, 0.f, 0.f);
        if (m < N_REL) x = *(const float4*)(metarel + m * DIM + col);
        *(float4*)(aw + r * A_STRIDE_N + col) = x;
    }

    for (int nt = 0; nt < 8; ++nt) {
        const int n0 = nt * 16;
        v8f acc = {};
#pragma unroll 4
        for (int ks = 0; ks < 32; ++ks) {
            const int ao = lr * A_STRIDE_N + ks * 4 + half * 2;
            v2f a;
            a.x = aw[ao];
            a.y = aw[ao + 1];
            const int bo0 = (ks * 4 + half * 2) * DIM + n0 + lr;
            v2f b;
            b.x = WtM[bo0];
            b.y = WtM[bo0 + DIM];
            acc = __builtin_amdgcn_wmma_f32_16x16x4_f32(false, a, false, b,
                                                        (short)0, acc, false, false);
        }
        const int n = n0 + lr;
        const float bias = bM[n];
#pragma unroll
        for (int r = 0; r < 8; ++r) {
            const int m = m0 + r + half * 8;
            if (m < N_REL) out[m * DIM + n] = acc[r] + bias;
        }
    }
}

// ---------------------------------------------------------------- launcher
extern "C" void kernel_launch(void* const* d_in, const int* in_sizes, int n_in,
                              void* d_out, int out_size, void* d_ws, size_t ws_size,
                              hipStream_t stream) {
    const float* rel_emb = (const float*)d_in[0];
    const float* metarel = (const float*)d_in[1];
    const int*   srcI    = (const int*)  d_in[2];
    const int*   dstI    = (const int*)  d_in[3];
    const int*   relI    = (const int*)  d_in[4];
    const int*   invI    = (const int*)  d_in[5];
    const float* W_I_w   = (const float*)d_in[6];
    const float* W_I_b   = (const float*)d_in[7];
    const float* W_O_w   = (const float*)d_in[8];
    const float* W_O_b   = (const float*)d_in[9];
    const float* W_S_w   = (const float*)d_in[10];
    const float* W_S_b   = (const float*)d_in[11];
    const float* W_M_w   = (const float*)d_in[12];
    const float* W_M_b   = (const float*)d_in[13];

    float* ws  = (float*)d_ws;
    float* WtI = ws + WS_WT_I;
    float* WtO = ws + WS_WT_O;
    float* WtS = ws + WS_WT_S;
    float* WtM = ws + WS_WT_M;
    float* agg = ws + WS_AGG;
    float* cnt = ws + WS_CNT;

    float* out_rel  = (float*)d_out;                       // 50000*128
    float* out_meta = (float*)d_out + N_NODES * DIM;       // 500*128

    const int edgeLds = EDGE_WAVES * EDGES_PER_WAVE * A_STRIDE_E * sizeof(float); // 266240 B
    const int nodeLds = 8 * 16 * A_STRIDE_N * sizeof(float);                      // 67584 B

    // allow >64KB dynamic LDS (host-side, graph-capture safe, deterministic)
    (void)hipFuncSetAttribute((const void*)k_edge,
                              hipFuncAttributeMaxDynamicSharedMemorySize, edgeLds);
    (void)hipFuncSetAttribute((const void*)k_node,
                              hipFuncAttributeMaxDynamicSharedMemorySize, nodeLds);
    (void)hipFuncSetAttribute((const void*)k_meta,
                              hipFuncAttributeMaxDynamicSharedMemorySize, nodeLds);

    // 1) zero accumulation scratch (agg + cnt) every call
    k_zero<<<(WS_ZERO_N + 255) / 256, 256, 0, stream>>>(agg, WS_ZERO_N);

    // 2) transpose weights: W[n][k] -> Wt[k][n]
    k_transpose<<<(DIM * K2 + 255) / 256, 256, 0, stream>>>(W_I_w, WtI, DIM, K2);
    k_transpose<<<(DIM * K2 + 255) / 256, 256, 0, stream>>>(W_O_w, WtO, DIM, K2);
    k_transpose<<<(DIM * DIM + 255) / 256, 256, 0, stream>>>(W_S_w, WtS, DIM, DIM);
    k_transpose<<<(DIM * DIM + 255) / 256, 256, 0, stream>>>(W_M_w, WtM, DIM, DIM);

    // 3) edge messages + scatter-add (WMMA f32, 4x M-register-blocked)
    k_edge<<<N_EDGES / EDGES_PER_BLOCK, 128, edgeLds, stream>>>(rel_emb, metarel,
                                                                srcI, dstI, relI, invI,
                                                                WtI, W_I_b, WtO, W_O_b,
                                                                agg, cnt);

    // 4) node update (WMMA f32 + mean-aggregation)
    k_node<<<(N_NODES + 127) / 128, 256, nodeLds, stream>>>(rel_emb, WtS, W_S_b,
                                                            agg, cnt, out_rel);

    // 5) meta-relation update (WMMA f32)
    k_meta<<<(N_REL + 127) / 128, 256, nodeLds, stream>>>(metarel, WtM, W_M_b,
                                                          out_meta);
}